// CompressJpeg_283467842062
// MI455X (gfx1250) — compile-verified
//
#include <hip/hip_runtime.h>
#include <hip/hip_bf16.h>

typedef float v2f __attribute__((ext_vector_type(2)));
typedef float v8f __attribute__((ext_vector_type(8)));

// One 16x16x16 fp32 matmul as 4 chained V_WMMA_F32_16X16X4_F32.
// a8[2c+j] = A[m][4c+2h+j]  (A operand: lane m=L%16 holds its row, h=L/16)
// b8[2c+j] = B[4c+2h+j][n]  (B operand: lane n=L%16 holds its column)
__device__ __forceinline__ v8f wmma_k16(const float* a8, const float* b8, v8f acc) {
#pragma unroll
    for (int c = 0; c < 4; ++c) {
        v2f a;  a.x  = a8[2 * c]; a.y  = a8[2 * c + 1];
        v2f bb; bb.x = b8[2 * c]; bb.y = b8[2 * c + 1];
        acc = __builtin_amdgcn_wmma_f32_16x16x4_f32(
            /*neg_a=*/false, a, /*neg_b=*/false, bb,
            /*c_mod=*/(short)0, acc, /*reuse_a=*/false, /*reuse_b=*/false);
    }
    return acc;
}

// Second matmul (D = Ad^T * W) B-operand built from matmul-1 output (C/D layout:
// lane holds rows v+8h of column n) + its xor-16 partner (rows v+8(1-h)).
__device__ __forceinline__ void build_b2(const float* w, const float* xw, int h, float* pb) {
    pb[0] = h ? xw[2] : w[0];   // c=0: rows {2h, 2h+1}
    pb[1] = h ? xw[3] : w[1];
    pb[2] = h ? xw[6] : w[4];   // c=1: rows {4+2h, 5+2h}
    pb[3] = h ? xw[7] : w[5];
    pb[4] = h ? w[2] : xw[0];   // c=2: rows {8+2h, 9+2h}
    pb[5] = h ? w[3] : xw[1];
    pb[6] = h ? w[6] : xw[4];   // c=3: rows {12+2h, 13+2h}
    pb[7] = h ? w[7] : xw[5];
}

__global__ __launch_bounds__(128) void CompressJpeg_kernel(
    const float* __restrict__ img,    // [16,3,1024,1024]
    const float* __restrict__ mat,    // [3,3]  ycbcr = rgb*255 @ mat + shift
    const float* __restrict__ shf,    // [3]
    const float* __restrict__ ytab,   // [8,8]
    const float* __restrict__ ctab,   // [8,8]
    const float* __restrict__ dct,    // [8,8,8,8]  dct[x,y,u,v]; A[x][u]=dct[x,0,u,0]
    const float* __restrict__ dsc,    // [8,8]
    const int*   __restrict__ facp,
    float*       __restrict__ out)
{
    __shared__ float lds[4 * 2048];           // per wave: cb[32][32], cr[32][32]

    const int lane   = threadIdx.x & 31;
    const int wib    = threadIdx.x >> 5;
    const int waveId = blockIdx.x * 4 + wib;
    const int b      = waveId >> 10;          // image index (16 images x 1024 tiles)
    const int tIdx   = waveId & 1023;
    const int R0     = (tIdx >> 5) << 5;      // 32x32 region origin
    const int C0     = (tIdx & 31) << 5;
    const int m      = lane & 15;             // row (A) / col (B,C,D) index
    const int h      = lane >> 4;             // half
    const int q8     = m & 7;                 // column within 8x8 block
    const float fac  = (float)facp[0];

    float* cbL = &lds[wib * 2048];
    float* crL = cbL + 1024;

    // Block-diagonal DCT basis, serves as both Ad^T A-operand and Ad B-operand.
    float adv[8];
#pragma unroll
    for (int c = 0; c < 4; ++c)
#pragma unroll
        for (int j = 0; j < 2; ++j) {
            const int row = 4 * c + 2 * h + j;                 // K index
            const bool on = ((row < 8) == (m < 8));            // block-diagonal
            adv[2 * c + j] = on ? dct[(row & 7) * 512 + (m & 7) * 8] : 0.0f;
        }

    // Per-lane fused scale/quant reciprocals (u = register index v, col = q8):
    // out = rintf(D * dsc[u,v] / (tab[u,v]*fac))  ->  one v_mul + v_rndne each.
    float invY[8], invC[8];
#pragma unroll
    for (int v = 0; v < 8; ++v) {
        const float s = dsc[v * 8 + q8];
        invY[v] = s / (ytab[v * 8 + q8] * fac);
        invC[v] = s / (ctab[v * 8 + q8] * fac);
    }

    const float m00 = mat[0], m10 = mat[3], m20 = mat[6];      // Y
    const float m01 = mat[1], m11 = mat[4], m21 = mat[7];      // Cb
    const float m02 = mat[2], m12 = mat[5], m22 = mat[8];      // Cr
    const float s0 = shf[0], s1 = shf[1], s2 = shf[2];

    const float* rp = img + (size_t)b * 3145728;
    const float* gp = rp + 1048576;
    const float* bp = gp + 1048576;

    // ---------------- Y: four 16x16 tiles (16 blocks) ----------------
#pragma unroll
    for (int t = 0; t < 4; ++t) {
        const int tr = ((t >> 1) << 4);
        const int tc = ((t & 1) << 4);
        const int gr = R0 + tr + m;
        float pa[8];
#pragma unroll
        for (int c = 0; c < 4; ++c) {
            const int col = 4 * c + 2 * h;
            const size_t off = (size_t)gr * 1024 + (C0 + tc + col);
            const float2 rv = *(const float2*)(rp + off);
            const float2 gv = *(const float2*)(gp + off);
            const float2 bv = *(const float2*)(bp + off);
#pragma unroll
            for (int j = 0; j < 2; ++j) {
                const float R = (j ? rv.y : rv.x) * 255.0f;
                const float G = (j ? gv.y : gv.x) * 255.0f;
                const float B = (j ? bv.y : bv.x) * 255.0f;
                pa[2 * c + j] = R * m00 + G * m10 + B * m20 + s0 - 128.0f;
                const int rr = tr + m, cc = tc + col + j;
                cbL[rr * 32 + cc] = R * m01 + G * m11 + B * m21 + s1;
                crL[rr * 32 + cc] = R * m02 + G * m12 + B * m22 + s2;
            }
        }
        v8f acc = {};
        acc = wmma_k16(pa, adv, acc);                          // W = P * Ad
        float w[8], xw[8];
#pragma unroll
        for (int v = 0; v < 8; ++v) { w[v] = acc[v]; xw[v] = __shfl_xor(w[v], 16, 32); }
        float pb[8];
        build_b2(w, xw, h, pb);
        v8f d = {};
        d = wmma_k16(adv, pb, d);                              // D = Ad^T * W

        const int hb = (R0 >> 3) + ((t >> 1) << 1) + h;
        const int wb = (C0 >> 3) + ((t & 1) << 1) + (m >> 3);
        const long base = ((long)b * 16384 + (long)hb * 128 + wb) * 64 + q8;
#pragma unroll
        for (int v = 0; v < 8; ++v)
            out[base + v * 8] = rintf(d[v] * invY[v]);
    }

    __syncthreads();

    // ---------------- Cb / Cr: one pooled 16x16 tile each (4 blocks) ----------------
#pragma unroll
    for (int chn = 0; chn < 2; ++chn) {
        const float* src = chn ? crL : cbL;
        float pa[8];
#pragma unroll
        for (int c = 0; c < 4; ++c)
#pragma unroll
            for (int j = 0; j < 2; ++j) {
                const int cc = 4 * c + 2 * h + j;
                const int r2 = m << 1, c2 = cc << 1;
                const float s = src[r2 * 32 + c2]       + src[r2 * 32 + c2 + 1] +
                                src[(r2 + 1) * 32 + c2] + src[(r2 + 1) * 32 + c2 + 1];
                pa[2 * c + j] = 0.25f * s - 128.0f;
            }
        v8f acc = {};
        acc = wmma_k16(pa, adv, acc);
        float w[8], xw[8];
#pragma unroll
        for (int v = 0; v < 8; ++v) { w[v] = acc[v]; xw[v] = __shfl_xor(w[v], 16, 32); }
        float pb[8];
        build_b2(w, xw, h, pb);
        v8f d = {};
        d = wmma_k16(adv, pb, d);

        const int hb = (R0 >> 4) + h;                          // chroma plane is 512x512
        const int wb = (C0 >> 4) + (m >> 3);
        const long base = 16777216L + (chn ? 4194304L : 0L) +
                          ((long)b * 4096 + (long)hb * 64 + wb) * 64 + q8;
#pragma unroll
        for (int v = 0; v < 8; ++v)
            out[base + v * 8] = rintf(d[v] * invC[v]);
    }
}

extern "C" void kernel_launch(void* const* d_in, const int* in_sizes, int n_in,
                              void* d_out, int out_size, void* d_ws, size_t ws_size,
                              hipStream_t stream) {
    const float* image  = (const float*)d_in[0];
    const float* matrix = (const float*)d_in[1];
    const float* shift  = (const float*)d_in[2];
    const float* ytab   = (const float*)d_in[3];
    const float* ctab   = (const float*)d_in[4];
    const float* dctt   = (const float*)d_in[5];
    const float* dscale = (const float*)d_in[6];
    const int*   factor = (const int*)d_in[7];
    float*       outp   = (float*)d_out;

    // 16 images * (1024/32)^2 = 16384 waves; 4 waves per 128-thread block.
    dim3 grid(4096), block(128);
    CompressJpeg_kernel<<<grid, block, 0, stream>>>(
        image, matrix, shift, ytab, ctab, dctt, dscale, factor, outp);
}